// WordCharEncoderBiRNN_42889543417938
// MI455X (gfx1250) — compile-verified
//
#include <hip/hip_runtime.h>
#include <hip/hip_bf16.h>
#include <math.h>

// ---------------- problem constants ----------------
#define L   256
#define D   512
#define Hh  1024
#define CL  16
#define C2c 172   // D/3 + D%3
#define C3c 170
#define C4c 170
#define XD  1024  // 2*D
#define G4  4096  // 4*H
#define NBLK_DIR 16   // workgroups per LSTM direction

typedef float v2f __attribute__((ext_vector_type(2)));
typedef float v8f __attribute__((ext_vector_type(8)));

// ---------------- init: zero barrier state ----------------
__global__ void init_bar_kernel(unsigned* bar) {
    if (threadIdx.x < 4) bar[threadIdx.x] = 0u;
}

// ---------------- char CNN + build x ----------------
// one block per word; emb (16x512) staged in LDS
__global__ __launch_bounds__(256)
void char_cnn_kernel(const float* __restrict__ word_emb,
                     const int*   __restrict__ char_idxs,
                     const float* __restrict__ char_embed,
                     const float* __restrict__ W2, const float* __restrict__ b2,
                     const float* __restrict__ W3, const float* __restrict__ b3,
                     const float* __restrict__ W4, const float* __restrict__ b4,
                     float* __restrict__ x) {
    __shared__ float emb[CL * D]; // 32KB
    const int w   = blockIdx.x;
    const int tid = threadIdx.x;

    for (int i = tid; i < CL * D; i += 256) {
        int t = i >> 9, d = i & 511;
        emb[i] = char_embed[(size_t)char_idxs[w * CL + t] * D + d];
    }
    for (int d = tid; d < D; d += 256)
        x[(size_t)w * XD + d] = word_emb[(size_t)w * D + d];
    __syncthreads();

    for (int task = tid; task < 512; task += 256) {
        int K, c, off;
        const float* W; const float* b;
        if (task < C2c)            { K = 2; c = task;             W = W2; b = b2; off = D; }
        else if (task < C2c + C3c) { K = 3; c = task - C2c;       W = W3; b = b3; off = D + C2c; }
        else                       { K = 4; c = task - C2c - C3c; W = W4; b = b4; off = D + C2c + C3c; }
        const int T = CL - K + 1;
        float best = 0.f; // max(relu(.)) >= 0
        for (int t = 0; t < T; ++t) {
            float s = b[c];
            for (int k = 0; k < K; ++k) {
                const float4* wr4 = (const float4*)(W + ((size_t)c * K + k) * D);
                const float4* er4 = (const float4*)(emb + (t + k) * D);
                float acc = 0.f;
                #pragma unroll 4
                for (int q = 0; q < D / 4; ++q) {
                    float4 a = er4[q], v = wr4[q];
                    acc += a.x * v.x + a.y * v.y + a.z * v.z + a.w * v.w;
                }
                s += acc;
            }
            float r = s > 0.f ? s : 0.f;
            best = r > best ? r : best;
        }
        x[(size_t)w * XD + off + c] = best;
    }
}

// ---------------- input projection GEMM via V_WMMA_F32_16X16X4_F32 ----------
// Z[m][n] = sum_k x[m][k]*Wih[n][k] + (bih[n]+bhh[n]);  M=256, N=4096, K=1024
// One wave computes one 16x16 tile.  Per-lane fragment layout (ISA 7.12.2):
//   A: VGPR r, lane l -> A[l%16][2*(l>>4)+r]   (16x4, f32)
//   B: VGPR r, lane l -> B[2*(l>>4)+r][l%16]   (4x16, f32)
//   C/D: VGPR r, lane l -> (M = r + 8*(l>>4), N = l%16)
__global__ __launch_bounds__(256)
void lstm_inproj_wmma(const float* __restrict__ x,    // 256 x 1024
                      const float* __restrict__ Wih,  // 4096 x 1024
                      const float* __restrict__ bih,
                      const float* __restrict__ bhh,
                      float* __restrict__ Z) {        // 256 x 4096
    const int wave  = threadIdx.x >> 5;
    const int lane  = threadIdx.x & 31;
    const int tile  = blockIdx.x * (blockDim.x >> 5) + wave;
    const int tilesN = G4 / 16;           // 256
    const int m0 = (tile / tilesN) * 16;
    const int n0 = (tile % tilesN) * 16;
    const int lmod = lane & 15;
    const int lhi  = lane >> 4;

    const float* arow = x   + (size_t)(m0 + lmod) * XD;
    const float* brow = Wih + (size_t)(n0 + lmod) * XD;

    v8f acc = {};
    for (int k = 0; k < XD; k += 4) {
        const int ka = k + 2 * lhi;                // even -> 8B aligned
        v2f a = *(const v2f*)(arow + ka);
        v2f b = *(const v2f*)(brow + ka);
        acc = __builtin_amdgcn_wmma_f32_16x16x4_f32(
                  false, a, false, b, (short)0, acc, false, false);
    }
    const int n = n0 + lmod;
    const float bias = bih[n] + bhh[n];
    #pragma unroll
    for (int r = 0; r < 8; ++r) {
        const int m = m0 + r + 8 * lhi;
        Z[(size_t)m * G4 + n] = acc[r] + bias;
    }
}

// ---------------- cooperative LSTM recurrence ----------------
// 32 blocks: blocks [0,16) = forward, [16,32) = reverse.
// Block g of a direction owns h-indices [g*64, (g+1)*64) -> 256 gate rows.
__device__ __forceinline__ float sigm(float v) { return 1.f / (1.f + expf(-v)); }

__global__ __launch_bounds__(256)
void lstm_rec_kernel(const float* __restrict__ Zf, const float* __restrict__ Zr,
                     const float* __restrict__ Whh_f, const float* __restrict__ Whh_r,
                     float* __restrict__ out,      // 256 x 2048
                     float* __restrict__ hc_out,   // 4096: [h_f h_r c_f c_r]
                     float* __restrict__ hglob,    // 2 x 1024
                     unsigned* __restrict__ bar) { // [0..1]=cnt, [2..3]=gen
    const int dir = blockIdx.x >> 4;
    const int g   = blockIdx.x & 15;
    const int tid = threadIdx.x;
    const float* Z   = dir ? Zr : Zf;
    const float* Whh = dir ? Whh_r : Whh_f;
    const int gate = tid >> 6;          // 0..3
    const int jj   = tid & 63;
    const int j    = g * 64 + jj;
    const int row  = gate * Hh + j;
    const float4* wrow = (const float4*)(Whh + (size_t)row * Hh);

    __shared__ float hprev[Hh];
    __shared__ float zsh[4][64];
    __shared__ float csh[64];

    for (int i = tid; i < Hh; i += 256) hprev[i] = 0.f;
    if (tid < 64) csh[tid] = 0.f;
    __syncthreads();

    unsigned gen_local = 0;
    unsigned* cnt = bar + dir;
    unsigned* gen = bar + 2 + dir;

    for (int step = 0; step < L; ++step) {
        const int t = dir ? (L - 1 - step) : step;

        const float4* hp = (const float4*)hprev;
        float acc = 0.f;
        #pragma unroll 4
        for (int q = 0; q < Hh / 4; ++q) {
            float4 wv = wrow[q], hv = hp[q];
            acc += wv.x * hv.x + wv.y * hv.y + wv.z * hv.z + wv.w * hv.w;
        }
        zsh[gate][jj] = Z[(size_t)t * G4 + row] + acc;
        __syncthreads();

        if (tid < 64) {
            const float zi = zsh[0][tid], zf = zsh[1][tid];
            const float zg = zsh[2][tid], zo = zsh[3][tid];
            float c = sigm(zf) * csh[tid] + sigm(zi) * tanhf(zg);
            csh[tid] = c;
            float h = sigm(zo) * tanhf(c);
            hglob[dir * Hh + g * 64 + tid] = h;
            out[(size_t)t * 2048 + dir * Hh + g * 64 + tid] = h;
        }
        __threadfence();
        __syncthreads();

        // per-direction grid barrier across NBLK_DIR blocks
        if (tid == 0) {
            const unsigned target = gen_local + 1;
            const unsigned old = atomicAdd(cnt, 1u);
            if (old == NBLK_DIR - 1) {
                atomicExch(cnt, 0u);
                __threadfence();
                atomicExch(gen, target);
            } else {
                while (atomicAdd(gen, 0u) < target) { __builtin_amdgcn_s_sleep(1); }
            }
        }
        gen_local++;
        __syncthreads();
        __threadfence();   // acquire: see other blocks' h slices

        for (int i = tid; i < Hh; i += 256) hprev[i] = hglob[dir * Hh + i];
        __syncthreads();
    }

    if (tid < 64) {
        hc_out[dir * Hh + g * 64 + tid]        = hglob[dir * Hh + g * 64 + tid]; // hidden
        hc_out[2048 + dir * Hh + g * 64 + tid] = csh[tid];                       // cell
    }
}

// ---------------- launch ----------------
extern "C" void kernel_launch(void* const* d_in, const int* in_sizes, int n_in,
                              void* d_out, int out_size, void* d_ws, size_t ws_size,
                              hipStream_t stream) {
    const float* word_emb  = (const float*)d_in[0];
    const int*   char_idxs = (const int*)  d_in[1];
    const float* char_emb  = (const float*)d_in[2];
    const float* W2 = (const float*)d_in[3];  const float* b2 = (const float*)d_in[4];
    const float* W3 = (const float*)d_in[5];  const float* b3 = (const float*)d_in[6];
    const float* W4 = (const float*)d_in[7];  const float* b4 = (const float*)d_in[8];
    const float* Wih_f = (const float*)d_in[9];
    const float* Whh_f = (const float*)d_in[10];
    const float* bih_f = (const float*)d_in[11];
    const float* bhh_f = (const float*)d_in[12];
    const float* Wih_r = (const float*)d_in[13];
    const float* Whh_r = (const float*)d_in[14];
    const float* bih_r = (const float*)d_in[15];
    const float* bhh_r = (const float*)d_in[16];

    // workspace layout
    char* ws = (char*)d_ws;
    float* x     = (float*)(ws);                               // 256*1024 f32 (1 MB)
    float* Zf    = (float*)(ws + (1u << 20));                  // 256*4096 f32 (4 MB)
    float* Zr    = (float*)(ws + (5u << 20));                  // 4 MB
    float* hglob = (float*)(ws + (9u << 20));                  // 2*1024 f32
    unsigned* bar = (unsigned*)(ws + (9u << 20) + 8192);       // 4 u32

    float* out    = (float*)d_out;            // 256 x 2048
    float* hc_out = out + (size_t)L * 2048;   // 2048 hidden + 2048 cell

    init_bar_kernel<<<1, 64, 0, stream>>>(bar);

    char_cnn_kernel<<<L, 256, 0, stream>>>(word_emb, char_idxs, char_emb,
                                           W2, b2, W3, b3, W4, b4, x);

    // 16 x 256 tiles = 4096 waves, 8 waves/block -> 512 blocks per direction
    lstm_inproj_wmma<<<512, 256, 0, stream>>>(x, Wih_f, bih_f, bhh_f, Zf);
    lstm_inproj_wmma<<<512, 256, 0, stream>>>(x, Wih_r, bih_r, bhh_r, Zr);

    lstm_rec_kernel<<<2 * NBLK_DIR, 256, 0, stream>>>(Zf, Zr, Whh_f, Whh_r,
                                                      out, hc_out, hglob, bar);
}